// SNR_iwae_71983651881559
// MI455X (gfx1250) — compile-verified
//
#include <hip/hip_runtime.h>

// Problem constants (match reference: N=1048576, D=64, BATCH=1024, K=1024).
#define NROWS 1048576
#define DDIM  64
#define BATCHN 1024
#define KSAMP 1024
#define LDA 68   // padded LDS stride (even -> 8B aligned b64 reads, bank-conflict free)

typedef __attribute__((ext_vector_type(2))) float v2f;
typedef __attribute__((ext_vector_type(8))) float v8f;

// ---------------------------------------------------------------------------
// Kernel 1: z = x@A + b via V_WMMA_F32_16X16X4_F32, fused with
//           lw[i] = -0.5*(||x_i - z_i||^2 + ||z_i - theta||^2)
// One wave per 16-row tile (grid-stride).
// ---------------------------------------------------------------------------
__global__ __launch_bounds__(256) void iwae_gemm_logw(
    const float* __restrict__ x, const float* __restrict__ A,
    const float* __restrict__ b, const float* __restrict__ theta,
    float* __restrict__ lw)
{
    // Stage A transposed into LDS: At[c*LDA + r] = A[r*DDIM + c]
    __shared__ float At[DDIM * LDA];
    for (int i = threadIdx.x; i < DDIM * DDIM; i += blockDim.x) {
        int r = i >> 6, c = i & 63;
        At[c * LDA + r] = A[i];
    }
    __syncthreads();

    const int lane = threadIdx.x & 31;
    const int l16  = lane & 15;
    const int half = lane >> 4;           // 0: K={0,1} / rows 0-7, 1: K={2,3} / rows 8-15

    const int wavesPerBlock = blockDim.x >> 5;
    const int waveGlobal    = blockIdx.x * wavesPerBlock + (threadIdx.x >> 5);
    const int numWaves      = gridDim.x * wavesPerBlock;

    // Per-column constants for the 4 N-tiles (col = 16n + l16)
    float bn[4], thn[4];
#pragma unroll
    for (int n = 0; n < 4; ++n) {
        bn[n]  = b[16 * n + l16];
        thn[n] = theta[16 * n + l16];
    }

    const int numTiles = NROWS / 16;
    for (int tile = waveGlobal; tile < numTiles; tile += numWaves) {
        const int m0 = tile * 16;

        // A-fragments of x: lane holds x[m0+l16][4kk+2*half .. +1]  (16x4 f32 layout)
        v2f xa[16];
        const float* xrow = x + (size_t)(m0 + l16) * DDIM + 2 * half;
#pragma unroll
        for (int kk = 0; kk < 16; ++kk)
            xa[kk] = *(const v2f*)(xrow + 4 * kk);

        float s[8];
#pragma unroll
        for (int v = 0; v < 8; ++v) s[v] = 0.0f;

#pragma unroll
        for (int n = 0; n < 4; ++n) {
            v8f acc = {};
#pragma unroll
            for (int kk = 0; kk < 16; ++kk) {
                // B-fragment: v0=A[4kk+2h][16n+l16], v1=A[4kk+2h+1][16n+l16]
                v2f bf = *(const v2f*)(&At[(16 * n + l16) * LDA + 4 * kk + 2 * half]);
                acc = __builtin_amdgcn_wmma_f32_16x16x4_f32(
                        false, xa[kk], false, bf, (short)0, acc, false, false);
            }
            // Consume this 16x16 tile of z: lane=col 16n+l16, VGPR v = row m0+v+8*half
            const int col = 16 * n + l16;
#pragma unroll
            for (int v = 0; v < 8; ++v) {
                const int row = m0 + v + 8 * half;
                float z  = acc[v] + bn[n];
                float xe = x[(size_t)row * DDIM + col];
                float d1 = xe - z;
                float d2 = z - thn[n];
                s[v] += d1 * d1 + d2 * d2;
            }
        }

        // Reduce each row sum across the 16 lanes of its half (xor masks stay in-half)
#pragma unroll
        for (int v = 0; v < 8; ++v) {
            float t = s[v];
            t += __shfl_xor(t, 1, 32);
            t += __shfl_xor(t, 2, 32);
            t += __shfl_xor(t, 4, 32);
            t += __shfl_xor(t, 8, 32);
            s[v] = -0.5f * t;
        }
        if (l16 == 0) {  // lane 0 -> rows m0..m0+7, lane 16 -> rows m0+8..m0+15
            float* dst = lw + m0 + 8 * half;
            *(float4*)(dst)     = make_float4(s[0], s[1], s[2], s[3]);
            *(float4*)(dst + 4) = make_float4(s[4], s[5], s[6], s[7]);
        }
    }
}

// ---------------------------------------------------------------------------
// Kernel 2: per-batch-row softmax-weighted sum: rowval[r] = sum_k w_k * lw_k
// One 256-thread block per row of [1024 x 1024].
// ---------------------------------------------------------------------------
__global__ __launch_bounds__(256) void iwae_row_softmax(
    const float* __restrict__ lw, float* __restrict__ rowval)
{
    const int row = blockIdx.x;
    const int tid = threadIdx.x;
    const float4 v = ((const float4*)(lw + (size_t)row * KSAMP))[tid];

    __shared__ float rm[8], r1[8], r2[8];

    // row max
    float m = fmaxf(fmaxf(v.x, v.y), fmaxf(v.z, v.w));
    for (int off = 16; off > 0; off >>= 1) m = fmaxf(m, __shfl_xor(m, off, 32));
    if ((tid & 31) == 0) rm[tid >> 5] = m;
    __syncthreads();
    m = rm[0];
#pragma unroll
    for (int i = 1; i < 8; ++i) m = fmaxf(m, rm[i]);
    __syncthreads();

    // sum exp and sum exp*lw
    float e0 = __expf(v.x - m), e1 = __expf(v.y - m);
    float e2 = __expf(v.z - m), e3 = __expf(v.w - m);
    float s1 = e0 + e1 + e2 + e3;
    float s2 = e0 * v.x + e1 * v.y + e2 * v.z + e3 * v.w;
    for (int off = 16; off > 0; off >>= 1) {
        s1 += __shfl_xor(s1, off, 32);
        s2 += __shfl_xor(s2, off, 32);
    }
    if ((tid & 31) == 0) { r1[tid >> 5] = s1; r2[tid >> 5] = s2; }
    __syncthreads();
    if (tid == 0) {
        float S1 = 0.f, S2 = 0.f;
#pragma unroll
        for (int i = 0; i < 8; ++i) { S1 += r1[i]; S2 += r2[i]; }
        rowval[row] = S2 / S1;
    }
}

// ---------------------------------------------------------------------------
// Kernel 3: ELBO = -(1/BATCH) * sum(rowval)   (single block -> deterministic)
// ---------------------------------------------------------------------------
__global__ __launch_bounds__(256) void iwae_final(
    const float* __restrict__ rowval, float* __restrict__ out)
{
    const int tid = threadIdx.x;
    const float4 v = ((const float4*)rowval)[tid];   // 256*4 = 1024
    float s = v.x + v.y + v.z + v.w;
    for (int off = 16; off > 0; off >>= 1) s += __shfl_xor(s, off, 32);
    __shared__ float r[8];
    if ((tid & 31) == 0) r[tid >> 5] = s;
    __syncthreads();
    if (tid == 0) {
        float S = 0.f;
#pragma unroll
        for (int i = 0; i < 8; ++i) S += r[i];
        out[0] = -S / (float)BATCHN;
    }
}

extern "C" void kernel_launch(void* const* d_in, const int* in_sizes, int n_in,
                              void* d_out, int out_size, void* d_ws, size_t ws_size,
                              hipStream_t stream) {
    (void)in_sizes; (void)n_in; (void)out_size; (void)ws_size;
    const float* x     = (const float*)d_in[0];
    const float* A     = (const float*)d_in[1];
    const float* b     = (const float*)d_in[2];
    const float* theta = (const float*)d_in[3];
    // d_in[4] is K (==1024), baked into KSAMP.

    float* lw     = (float*)d_ws;        // N floats = 4 MB
    float* rowval = lw + NROWS;          // 1024 floats
    float* out    = (float*)d_out;

    // 1024 blocks x 8 waves = 8192 waves; 65536 tiles -> 8 tiles/wave
    iwae_gemm_logw<<<1024, 256, 0, stream>>>(x, A, b, theta, lw);
    iwae_row_softmax<<<BATCHN, 256, 0, stream>>>(lw, rowval);
    iwae_final<<<1, 256, 0, stream>>>(rowval, out);
}